// LinearAttentionMemory_12326556139956
// MI455X (gfx1250) — compile-verified
//
#include <hip/hip_runtime.h>

// ---------------------------------------------------------------------------
// LinearAttentionMemory cost scan, reformulated as prefix sums + per-token
// 64x64x64 f32 GEMM on V_WMMA_F32_16X16X4_F32 (CDNA5 / gfx1250, wave32).
//
// cost_l = 0.5*S_l/l - ||Kv_l||^2/l^2 + 0.5*<Kk_l@Kv_l, Kv_l>/l^3
//   Kv_l = sum_{i<=l} k_i v_i^T, Kk_l = sum k_i k_i^T, S_l = sum ||v_i||^2
// (q is dead: the reference's update_correlations call swaps its args; and
//  J == Skv^T bit-exactly, Svv only enters via its trace.)
//
// Kk is bit-exactly symmetric, so WMMA A-fragments are loaded row-major
// (A[m][kk] = Kk[kk][m]) -> conflict-free LDS access like the B side.
// ---------------------------------------------------------------------------

typedef __attribute__((ext_vector_type(2))) float v2f;   // 16x4 / 4x16 f32 fragment
typedef __attribute__((ext_vector_type(8))) float v8f;   // 16x16 f32 accumulator

#define DKV   64
#define BATCH 8
#define SEQ   2048
#define CHUNK 128
#define NCH   (SEQ / CHUNK)   // 16
#define SUB   32              // tokens staged per LDS pass in kernel 1

static __device__ __forceinline__ v8f wmma_f32_4(v2f a, v2f b, v8f c) {
  return __builtin_amdgcn_wmma_f32_16x16x4_f32(
      /*neg_a=*/false, a, /*neg_b=*/false, b,
      /*c_mod=*/(short)0, c, /*reuse_a=*/false, /*reuse_b=*/false);
}

// ---------------------------------------------------------------------------
// Kernel 1: per-(stream, chunk) sums  Ckv = K^T V,  Ckk = K^T K,  cvv = sum v^2
// One workgroup (8 wave32) per chunk; GEMM over the 128-token K dimension.
// Both output tiles of a wave share a tile-row -> one A fragment per k-step.
// ---------------------------------------------------------------------------
__global__ __launch_bounds__(256)
void la_chunk_sums(const float* __restrict__ kin,
                   const float* __restrict__ vin,
                   float* __restrict__ cKV,
                   float* __restrict__ cKK,
                   float* __restrict__ cVV) {
  __shared__ float sK[SUB * DKV];
  __shared__ float sV[SUB * DKV];
  __shared__ float sVVacc;

  const int blk  = blockIdx.x;            // b * NCH + ch
  const int b    = blk / NCH;
  const int ch   = blk % NCH;
  const int tid  = threadIdx.x;
  const int lane = tid & 31;
  const int wv   = tid >> 5;              // wave 0..7
  const int half = lane >> 4;             // lane group 0/1
  const int lm   = lane & 15;

  if (tid == 0) sVVacc = 0.0f;

  // two 16x16 output tiles per wave (16 tiles total); wv*2 even => same tm
  const int t0 = wv * 2, t1 = t0 + 1;
  const int tm  = t0 >> 2;                // shared tile row
  const int tn0 = t0 & 3, tn1 = t1 & 3;
  v8f accKV0 = {}, accKV1 = {}, accKK0 = {}, accKK1 = {};

  const float* kg = kin + ((size_t)b * SEQ + (size_t)ch * CHUNK) * DKV;
  const float* vg = vin + ((size_t)b * SEQ + (size_t)ch * CHUNK) * DKV;

  for (int sc = 0; sc < CHUNK; sc += SUB) {
    __syncthreads();
    // stage SUB tokens of K and V (2048 floats each), accumulate ||v||^2
    float vv = 0.0f;
    for (int e = tid; e < SUB * DKV; e += 256) {
      sK[e] = kg[(size_t)sc * DKV + e];
      float x = vg[(size_t)sc * DKV + e];
      sV[e] = x;
      vv += x * x;
    }
    atomicAdd(&sVVacc, vv);               // ds_add_f32
    __syncthreads();

    // GEMM over staged tokens, 4 at a time:  A[m][t] = K[t][m], B[t][n] = V/K[t][n]
    for (int kk = 0; kk < SUB; kk += 4) {
      const int kr = kk + half * 2;
      v2f aT, bV0, bV1, bK0, bK1;
      aT[0]  = sK[(kr + 0) * DKV + tm * 16 + lm];    // A fragment (shared by both tiles)
      aT[1]  = sK[(kr + 1) * DKV + tm * 16 + lm];
      bV0[0] = sV[(kr + 0) * DKV + tn0 * 16 + lm];   // B from V, tile col tn0
      bV0[1] = sV[(kr + 1) * DKV + tn0 * 16 + lm];
      bV1[0] = sV[(kr + 0) * DKV + tn1 * 16 + lm];
      bV1[1] = sV[(kr + 1) * DKV + tn1 * 16 + lm];
      bK0[0] = sK[(kr + 0) * DKV + tn0 * 16 + lm];   // B from K (for K^T K)
      bK0[1] = sK[(kr + 1) * DKV + tn0 * 16 + lm];
      bK1[0] = sK[(kr + 0) * DKV + tn1 * 16 + lm];
      bK1[1] = sK[(kr + 1) * DKV + tn1 * 16 + lm];

      accKV0 = wmma_f32_4(aT, bV0, accKV0);
      accKV1 = wmma_f32_4(aT, bV1, accKV1);
      accKK0 = wmma_f32_4(aT, bK0, accKK0);
      accKK1 = wmma_f32_4(aT, bK1, accKK1);
    }
  }

  // write 16x16 tiles: D VGPR r holds rows r (lanes 0-15) and r+8 (lanes 16-31)
  float* oKV = cKV + (size_t)blk * (DKV * DKV);
  float* oKK = cKK + (size_t)blk * (DKV * DKV);
#pragma unroll
  for (int r = 0; r < 8; ++r) {
    const int m  = tm * 16 + r + half * 8;
    const int n0 = tn0 * 16 + lm, n1 = tn1 * 16 + lm;
    oKV[m * DKV + n0] = accKV0[r];
    oKK[m * DKV + n0] = accKK0[r];
    oKV[m * DKV + n1] = accKV1[r];
    oKK[m * DKV + n1] = accKK1[r];
  }
  __syncthreads();
  if (tid == 0) cVV[blk] = sVVacc;
}

// ---------------------------------------------------------------------------
// Kernel 2: in-place exclusive prefix over the NCH chunk sums, per stream.
// Each thread owns 16 fixed elements of the 64x64 matrices.
// ---------------------------------------------------------------------------
__global__ __launch_bounds__(256)
void la_chunk_prefix(float* __restrict__ cKV,
                     float* __restrict__ cKK,
                     float* __restrict__ cVV) {
  const int b   = blockIdx.x;
  const int tid = threadIdx.x;
  float runKV[16], runKK[16];
#pragma unroll
  for (int i = 0; i < 16; ++i) { runKV[i] = 0.0f; runKK[i] = 0.0f; }

  for (int ch = 0; ch < NCH; ++ch) {
    float* pKV = cKV + ((size_t)b * NCH + ch) * (DKV * DKV);
    float* pKK = cKK + ((size_t)b * NCH + ch) * (DKV * DKV);
#pragma unroll
    for (int i = 0; i < 16; ++i) {
      const int e = i * 256 + tid;                  // coalesced
      float t = pKV[e]; pKV[e] = runKV[i]; runKV[i] += t;
      float u = pKK[e]; pKK[e] = runKK[i]; runKK[i] += u;
    }
  }
  if (tid == 0) {
    float r = 0.0f;
    for (int ch = 0; ch < NCH; ++ch) {
      float t = cVV[b * NCH + ch]; cVV[b * NCH + ch] = r; r += t;
    }
  }
}

// ---------------------------------------------------------------------------
// Kernel 3: per-(stream, chunk) 128-step scan. State Kv, Kk lives in LDS.
// Each step: rank-1 updates, then M = Kk @ Kv via 256 WMMA f32 16x16x4 ops
// spread over 8 waves, then the three trace terms via LDS-atomic reductions.
// A fragments exploit Kk symmetry -> row-major, bank-conflict-free loads.
// ---------------------------------------------------------------------------
__global__ __launch_bounds__(256)
void la_scan_cost(const float* __restrict__ kin,
                  const float* __restrict__ vin,
                  const float* __restrict__ cKV,
                  const float* __restrict__ cKK,
                  const float* __restrict__ cVV,
                  float* __restrict__ out) {
  __shared__ float sKV[DKV * DKV];
  __shared__ float sKK[DKV * DKV];
  __shared__ float skt[DKV];
  __shared__ float svt[DKV];
  __shared__ float red[4];   // [0]=||v_t||^2, [1]=||Kv||^2, [2]=<KkKv,Kv>, [3]=running S

  const int blk  = blockIdx.x;
  const int b    = blk / NCH;
  const int ch   = blk % NCH;
  const int tid  = threadIdx.x;
  const int lane = tid & 31;
  const int wv   = tid >> 5;
  const int half = lane >> 4;
  const int lm   = lane & 15;

  // load chunk-start exclusive-prefix state into LDS
  const float* pKV = cKV + (size_t)blk * (DKV * DKV);
  const float* pKK = cKK + (size_t)blk * (DKV * DKV);
  for (int i = tid; i < DKV * DKV; i += 256) { sKV[i] = pKV[i]; sKK[i] = pKK[i]; }
  if (tid == 0) { red[0] = 0.0f; red[1] = 0.0f; red[2] = 0.0f; red[3] = cVV[blk]; }

  const float* kg = kin + ((size_t)b * SEQ + (size_t)ch * CHUNK) * DKV;
  const float* vg = vin + ((size_t)b * SEQ + (size_t)ch * CHUNK) * DKV;
  float*       og = out + (size_t)b * SEQ + (size_t)ch * CHUNK;

  const int t0 = wv * 2, t1 = t0 + 1;
  const int tm  = t0 >> 2;                // shared tile row for both tiles
  const int tn0 = t0 & 3, tn1 = t1 & 3;

  for (int t = 0; t < CHUNK; ++t) {
    if (t + 1 < CHUNK && (tid & 63) == 0) {         // global_prefetch_b8 next token
      __builtin_prefetch(kg + (size_t)(t + 1) * DKV, 0, 3);
      __builtin_prefetch(vg + (size_t)(t + 1) * DKV, 0, 3);
    }
    __syncthreads();                                // red[] zeros visible

    // phase A: bring k_t, v_t into LDS; accumulate ||v_t||^2
    if (tid < DKV) {
      skt[tid] = kg[(size_t)t * DKV + tid];
    } else if (tid < 2 * DKV) {
      float x = vg[(size_t)t * DKV + (tid - DKV)];
      svt[tid - DKV] = x;
      atomicAdd(&red[0], x * x);
    }
    __syncthreads();

    // phase B: rank-1 updates Kv += k v^T, Kk += k k^T; accumulate ||Kv||^2
    {
      float t2p = 0.0f;
#pragma unroll
      for (int i = 0; i < 16; ++i) {
        const int e = i * 256 + tid;
        const int a = e >> 6, c = e & 63;
        float kv = sKV[e] + skt[a] * svt[c];
        sKV[e] = kv;
        sKK[e] += skt[a] * skt[c];
        t2p += kv * kv;
      }
      atomicAdd(&red[1], t2p);
    }
    __syncthreads();

    // phase C: M = Kk @ Kv via WMMA; accumulate <M, Kv> elementwise.
    // A[m][kk] = Kk[m][kk] = Kk[kk][m] (symmetry) -> row-major, conflict-free.
    {
      v8f acc0 = {}, acc1 = {};
#pragma unroll
      for (int kk = 0; kk < DKV; kk += 4) {
        const int kc = kk + half * 2;
        v2f a, b0, b1;
        a[0]  = sKK[(kc + 0) * DKV + tm * 16 + lm];   // shared A fragment
        a[1]  = sKK[(kc + 1) * DKV + tm * 16 + lm];
        b0[0] = sKV[(kc + 0) * DKV + tn0 * 16 + lm];
        b0[1] = sKV[(kc + 1) * DKV + tn0 * 16 + lm];
        b1[0] = sKV[(kc + 0) * DKV + tn1 * 16 + lm];
        b1[1] = sKV[(kc + 1) * DKV + tn1 * 16 + lm];
        acc0 = wmma_f32_4(a, b0, acc0);
        acc1 = wmma_f32_4(a, b1, acc1);
      }
      float t3p = 0.0f;
#pragma unroll
      for (int r = 0; r < 8; ++r) {
        const int m  = tm * 16 + r + half * 8;
        t3p += acc0[r] * sKV[m * DKV + tn0 * 16 + lm];
        t3p += acc1[r] * sKV[m * DKV + tn1 * 16 + lm];
      }
      atomicAdd(&red[2], t3p);
    }
    __syncthreads();

    // phase D: combine the three terms, emit cost, reset reduction slots
    if (tid == 0) {
      const float svv = red[3] + red[0];
      red[3] = svv;
      const float li  = (float)(ch * CHUNK + t + 1);
      const float il  = 1.0f / li;
      og[t] = 0.5f * svv * il - red[1] * il * il + 0.5f * red[2] * il * il * il;
      red[0] = 0.0f; red[1] = 0.0f; red[2] = 0.0f;
    }
  }
}

// ---------------------------------------------------------------------------
extern "C" void kernel_launch(void* const* d_in, const int* in_sizes, int n_in,
                              void* d_out, int out_size, void* d_ws, size_t ws_size,
                              hipStream_t stream) {
  (void)in_sizes; (void)n_in; (void)out_size; (void)ws_size;
  const float* q = (const float*)d_in[0];   (void)q;   // dead input (ref swaps args)
  const float* k = (const float*)d_in[1];
  const float* v = (const float*)d_in[2];
  float* out = (float*)d_out;

  float* ws  = (float*)d_ws;
  float* cKV = ws;                                         // B*NCH*4096 floats
  float* cKK = cKV + (size_t)BATCH * NCH * DKV * DKV;      // B*NCH*4096 floats
  float* cVV = cKK + (size_t)BATCH * NCH * DKV * DKV;      // B*NCH floats

  la_chunk_sums  <<<BATCH * NCH, 256, 0, stream>>>(k, v, cKV, cKK, cVV);
  la_chunk_prefix<<<BATCH,       256, 0, stream>>>(cKV, cKK, cVV);
  la_scan_cost   <<<BATCH * NCH, 256, 0, stream>>>(k, v, cKV, cKK, cVV, out);
}